// Loss_calculater_19610820673591
// MI455X (gfx1250) — compile-verified
//
#include <hip/hip_runtime.h>
#include <hip/hip_bf16.h>

#define IMG 640
#define A_ANCH 76725
#define NUM_CLASSES 80
#define MAX_GT 32
#define BATCH 8
#define POS_THR 0.5f
#define NEG_THR 0.4f
#define F_ALPHA 0.25f
#define F_BETA (1.0f / 9.0f)

typedef float v2f __attribute__((ext_vector_type(2)));
typedef float v8f __attribute__((ext_vector_type(8)));

// ---------------------------------------------------------------------------
// Wave (32-lane) sum reduction. On gfx1250, use V_WMMA_F32_16X16X4_F32:
//   A[m][0] = p[lane m] (lanes 0-15, VGPR0), A[m][2] = p[lane m+16], A[*][1,3]=0
//   B = ones(4x16)  =>  D[m][n] = p[m] + p[m+16]  for every column n.
// Sum of the 8 D VGPRs in lane n gives sum_{m=0..7}(p[m]+p[m+16]) (lanes 0-15)
// or sum_{m=8..15}(...) (lanes 16-31); one shfl_xor(16) completes the sum.
// Exact f32 arithmetic, EXEC must be all-ones at the call site.
// ---------------------------------------------------------------------------
__device__ __forceinline__ float wave_reduce_sum(float v) {
#if defined(__gfx1250__) && __has_builtin(__builtin_amdgcn_wmma_f32_16x16x4_f32)
  v2f a;
  a.x = v;
  a.y = 0.0f;
  v2f b;
  b.x = 1.0f;
  b.y = 1.0f;
  v8f c = {};
  c = __builtin_amdgcn_wmma_f32_16x16x4_f32(
      /*neg_a=*/false, a, /*neg_b=*/false, b,
      /*c_mod=*/(short)0, c, /*reuse_a=*/false, /*reuse_b=*/false);
  float t = ((c[0] + c[1]) + (c[2] + c[3])) + ((c[4] + c[5]) + (c[6] + c[7]));
  t += __shfl_xor(t, 16, 32);
  return t;
#else
  for (int off = 16; off > 0; off >>= 1) v += __shfl_xor(v, off, 32);
  return v;
#endif
}

// Block sum reduction (blockDim.x multiple of 32). Result valid in thread 0.
__device__ __forceinline__ float block_reduce_sum(float v, float* sbuf) {
  __syncthreads();  // allow sbuf reuse across successive calls
  const int lane = threadIdx.x & 31;
  const int wid = threadIdx.x >> 5;
  const int nw = blockDim.x >> 5;
  float w = wave_reduce_sum(v);
  if (lane == 0) sbuf[wid] = w;
  __syncthreads();
  float r = 0.0f;
  if (wid == 0) {
    float x = (lane < nw) ? sbuf[lane] : 0.0f;
    for (int off = 16; off > 0; off >>= 1) x += __shfl_xor(x, off, 32);
    r = x;
  }
  return r;
}

__device__ __forceinline__ float iou_one(float4 an, float area_a, float gx1,
                                         float gy1, float gx2, float gy2) {
  float ltx = fmaxf(an.x, gx1), lty = fmaxf(an.y, gy1);
  float rbx = fminf(an.z, gx2), rby = fminf(an.w, gy2);
  float w = fmaxf(rbx - ltx, 0.0f), h = fmaxf(rby - lty, 0.0f);
  float inter = w * h;
  float area_g = (gx2 - gx1) * (gy2 - gy1);
  return inter / (area_a + area_g - inter + 1e-6f);
}

// ---------------------------------------------------------------------------
// Kernel 0: zero accumulators and gt_max bits.
// ws layout:  [0..3] floats: reg_sum, cls_sum, npos, pad
//             +64B  : 256 uints  gt_max bits (B x MAX_GT)
//             +1088B: B*A bytes  per-anchor codes (label | 80=neg | 255=ignore)
// ---------------------------------------------------------------------------
__global__ void k_init(float* accum, unsigned int* gtm) {
  int i = threadIdx.x;
  if (i < 4) accum[i] = 0.0f;
  if (i < BATCH * MAX_GT) gtm[i] = 0u;
}

// ---------------------------------------------------------------------------
// Kernel 1: gt_max[b][g] = max over anchors of IoU (stored as nonneg f32 bits,
// so uint atomicMax ordering matches float ordering; is_best also requires
// gt_max > 0, so clamping negatives to 0 is equivalent).
// ---------------------------------------------------------------------------
__global__ void k_gtmax(const float* __restrict__ anchors,
                        const float* __restrict__ targets,
                        unsigned int* __restrict__ gtm) {
  __shared__ float sg[MAX_GT * 5];
  __shared__ unsigned int smax[MAX_GT];
  const int b = blockIdx.y;
  const int a = blockIdx.x * blockDim.x + threadIdx.x;
  if (threadIdx.x < MAX_GT * 5) sg[threadIdx.x] = targets[b * MAX_GT * 5 + threadIdx.x];
  if (threadIdx.x < MAX_GT) smax[threadIdx.x] = 0u;
  __syncthreads();
  if (a < A_ANCH) {
    float4 an = ((const float4*)anchors)[a];
    float area_a = (an.z - an.x) * (an.w - an.y);
#pragma unroll 4
    for (int g = 0; g < MAX_GT; ++g) {
      float iou = iou_one(an, area_a, sg[g * 5 + 0], sg[g * 5 + 1],
                          sg[g * 5 + 2], sg[g * 5 + 3]);
      atomicMax(&smax[g], __float_as_uint(fmaxf(iou, 0.0f)));
    }
  }
  __syncthreads();
  if (threadIdx.x < MAX_GT)
    atomicMax(&gtm[b * MAX_GT + threadIdx.x], smax[threadIdx.x]);
}

// ---------------------------------------------------------------------------
// Kernel 2: per-anchor assignment + smooth-L1 reg loss + npos.
// ---------------------------------------------------------------------------
__global__ void k_assign(const float* __restrict__ anchors,
                         const float* __restrict__ targets,
                         const float* __restrict__ reg_pred,
                         const unsigned int* __restrict__ gtm,
                         unsigned char* __restrict__ codes,
                         float* __restrict__ accum) {
  __shared__ float sg[MAX_GT * 5];
  __shared__ float sgm[MAX_GT];
  __shared__ float sred[8];
  const int b = blockIdx.y;
  const int a = blockIdx.x * blockDim.x + threadIdx.x;
  if (threadIdx.x < MAX_GT * 5) sg[threadIdx.x] = targets[b * MAX_GT * 5 + threadIdx.x];
  if (threadIdx.x < MAX_GT)
    sgm[threadIdx.x] = __uint_as_float(gtm[b * MAX_GT + threadIdx.x]);
  __syncthreads();

  float reg_contrib = 0.0f, pos_contrib = 0.0f;
  if (a < A_ANCH) {
    float4 an = ((const float4*)anchors)[a];
    float area_a = (an.z - an.x) * (an.w - an.y);
    float max_iou = -2.0f;
    int arg = 0;
    float best_v = -__builtin_inff();
    int best_g = -1;
#pragma unroll 4
    for (int g = 0; g < MAX_GT; ++g) {
      float iou = iou_one(an, area_a, sg[g * 5 + 0], sg[g * 5 + 1],
                          sg[g * 5 + 2], sg[g * 5 + 3]);
      if (iou > max_iou) { max_iou = iou; arg = g; }  // first-max (jnp.argmax)
      float gm = sgm[g];
      bool isb = (iou >= gm - 1e-5f) && (gm > 0.0f);
      if (isb && iou > best_v) { best_v = iou; best_g = g; }
    }
    int assigned;
    if (best_g >= 0)              assigned = best_g + 1;   // best-anchor override
    else if (max_iou >= POS_THR)  assigned = arg + 1;
    else if (max_iou < NEG_THR)   assigned = 0;
    else                          assigned = -1;

    unsigned char code;
    if (assigned > 0) {
      const int g = assigned - 1;
      code = (unsigned char)(int)sg[g * 5 + 4];
      float aw = an.z - an.x, ah = an.w - an.y;
      float ax = an.x + 0.5f * aw, ay = an.y + 0.5f * ah;
      float gw = fmaxf(sg[g * 5 + 2] - sg[g * 5 + 0], 1.0f);
      float gh = fmaxf(sg[g * 5 + 3] - sg[g * 5 + 1], 1.0f);
      float gx = sg[g * 5 + 0] + 0.5f * gw;
      float gy = sg[g * 5 + 1] + 0.5f * gh;
      float d0 = (gx - ax) / aw;
      float d1 = (gy - ay) / ah;
      float d2 = logf(gw / aw);
      float d3 = logf(gh / ah);
      float4 rp = ((const float4*)reg_pred)[(size_t)b * A_ANCH + a];
      float dd[4] = {fabsf(rp.x - d0), fabsf(rp.y - d1), fabsf(rp.z - d2),
                     fabsf(rp.w - d3)};
      float s = 0.0f;
#pragma unroll
      for (int i = 0; i < 4; ++i) {
        float d = dd[i];
        s += (d < F_BETA) ? (0.5f * d * d / F_BETA) : (d - 0.5f * F_BETA);
      }
      reg_contrib = s;
      pos_contrib = 1.0f;
    } else if (assigned == 0) {
      code = (unsigned char)NUM_CLASSES;  // negative: one-hot of class 80 == 0
    } else {
      code = 255;  // ignore: cls_w = 0
    }
    codes[(size_t)b * A_ANCH + a] = code;
  }
  float rs = block_reduce_sum(reg_contrib, sred);
  float ps = block_reduce_sum(pos_contrib, sred);
  if (threadIdx.x == 0) {
    atomicAdd(&accum[0], rs);
    atomicAdd(&accum[2], ps);
  }
}

// ---------------------------------------------------------------------------
// Kernel 3: focal loss over 8*76725*80 logits, float4-vectorized grid-stride
// stream (memory bound: ~196 MB @ 23.3 TB/s).
// ---------------------------------------------------------------------------
__global__ void k_focal(const float* __restrict__ cls_pred,
                        const unsigned char* __restrict__ codes,
                        float* __restrict__ accum) {
  __shared__ float sred[8];
  const float4* cp = (const float4*)cls_pred;
  const size_t N4 = (size_t)BATCH * A_ANCH * (NUM_CLASSES / 4);
  const size_t stride = (size_t)gridDim.x * blockDim.x;
  float acc = 0.0f;
  for (size_t f = (size_t)blockIdx.x * blockDim.x + threadIdx.x; f < N4;
       f += stride) {
    size_t anchor = f / (NUM_CLASSES / 4);
    int cb = (int)(f % (NUM_CLASSES / 4)) * 4;
    int code = codes[anchor];
    if (code != 255) {
      float4 x4 = cp[f];
      float xs[4] = {x4.x, x4.y, x4.z, x4.w};
#pragma unroll
      for (int i = 0; i < 4; ++i) {
        float x = xs[i];
        float t = (code == (cb + i)) ? 1.0f : 0.0f;
        float ce = fmaxf(x, 0.0f) - x * t + log1pf(expf(-fabsf(x)));
        float p = 1.0f / (1.0f + expf(-x));
        float pt = p * t + (1.0f - p) * (1.0f - t);
        float fw = F_ALPHA * t + (1.0f - F_ALPHA) * (1.0f - t);
        float om = 1.0f - pt;
        acc += fw * om * om * ce;
      }
    }
  }
  float s = block_reduce_sum(acc, sred);
  if (threadIdx.x == 0) atomicAdd(&accum[1], s);
}

// ---------------------------------------------------------------------------
// Kernel 4: finalize 3 scalars.
// ---------------------------------------------------------------------------
__global__ void k_final(const float* __restrict__ accum, float* __restrict__ out) {
  if (threadIdx.x == 0) {
    float np = fmaxf(accum[2], 1.0f);
    float lr = accum[0] / np;
    float lc = accum[1] / np;
    out[0] = lr + lc;
    out[1] = lr;
    out[2] = lc;
  }
}

extern "C" void kernel_launch(void* const* d_in, const int* in_sizes, int n_in,
                              void* d_out, int out_size, void* d_ws,
                              size_t ws_size, hipStream_t stream) {
  // Input order: imgs(unused), proposals_reg, proposals_cls, targets, anchors.
  const float* reg_pred = (const float*)d_in[1];
  const float* cls_pred = (const float*)d_in[2];
  const float* targets = (const float*)d_in[3];
  const float* anchors = (const float*)d_in[4];
  float* out = (float*)d_out;

  float* accum = (float*)d_ws;                                       // 4 f32
  unsigned int* gtm = (unsigned int*)((char*)d_ws + 64);             // 256 u32
  unsigned char* codes = (unsigned char*)((char*)d_ws + 64 + 1024);  // B*A u8

  k_init<<<1, 256, 0, stream>>>(accum, gtm);
  dim3 grid((A_ANCH + 255) / 256, BATCH);
  k_gtmax<<<grid, 256, 0, stream>>>(anchors, targets, gtm);
  k_assign<<<grid, 256, 0, stream>>>(anchors, targets, reg_pred, gtm, codes,
                                     accum);
  k_focal<<<2048, 256, 0, stream>>>(cls_pred, codes, accum);
  k_final<<<1, 32, 0, stream>>>(accum, out);
}